// NNUE_17454747091333
// MI455X (gfx1250) — compile-verified
//
#include <hip/hip_runtime.h>
#include <hip/hip_bf16.h>

// ---------------------------------------------------------------------------
// NNUE forward for MI455X (gfx1250, wave32).
//  Kernel 1: embedding-bag gather (affine + factored) + bias + pov blend + relu
//            -> base[N,1536] in workspace.  Gather-bound; affine table (151MB)
//            is L2-resident on MI455X (192MB L2), so this runs at L2 BW.
//  Kernel 2: fc0 GEMM [16,1536]x[1536,8] per wave using V_WMMA_F32_16X16X4_F32
//            (full-precision f32 matrix path), then tiny fc1/fc2/fc3 head.
// ---------------------------------------------------------------------------

#define NN    4096
#define KACT  32
#define BASE  768
#define TWOB  1536

typedef __attribute__((ext_vector_type(2))) float v2f;
typedef __attribute__((ext_vector_type(8))) float v8f;

// ------------------------------ Kernel 1 -----------------------------------
// grid = N blocks, 192 threads (6 waves); thread t owns columns [4t, 4t+4).
__global__ __launch_bounds__(192) void nnue_ft_blend(
    const float* __restrict__ pov,
    const int*   __restrict__ widx,
    const int*   __restrict__ bidx,
    const float* __restrict__ affine_w,
    const float* __restrict__ affine_b,
    const float* __restrict__ factored_w,
    float*       __restrict__ base)
{
  __shared__ int iw[KACT];
  __shared__ int ib[KACT];
  const int n = blockIdx.x;
  const int t = threadIdx.x;
  if (t < KACT)            iw[t]        = widx[n * KACT + t];
  else if (t < 2 * KACT)   ib[t - KACT] = bidx[n * KACT + (t - KACT)];
  __syncthreads();

  const int c = t * 4;
  float awx = 0.f, awy = 0.f, awz = 0.f, aww = 0.f;  // white features
  float abx = 0.f, aby = 0.f, abz = 0.f, abw = 0.f;  // black features

  for (int k = 0; k < KACT; ++k) {
    const int rw = iw[k];
    const int rb = ib[k];
    if (k + 1 < KACT) {
      // gfx1250 global_prefetch_b8 for next gather rows (L2 hit warming)
      __builtin_prefetch(&affine_w[(size_t)iw[k + 1] * BASE + c], 0, 1);
      __builtin_prefetch(&affine_w[(size_t)ib[k + 1] * BASE + c], 0, 1);
    }
    const float4 w1 = *(const float4*)&affine_w  [(size_t)rw * BASE + c];
    const float4 w2 = *(const float4*)&factored_w[(size_t)(rw % BASE) * BASE + c];
    const float4 b1 = *(const float4*)&affine_w  [(size_t)rb * BASE + c];
    const float4 b2 = *(const float4*)&factored_w[(size_t)(rb % BASE) * BASE + c];
    awx += w1.x + w2.x; awy += w1.y + w2.y; awz += w1.z + w2.z; aww += w1.w + w2.w;
    abx += b1.x + b2.x; aby += b1.y + b2.y; abz += b1.z + b2.z; abw += b1.w + b2.w;
  }

  const float4 bias = *(const float4*)&affine_b[c];
  awx += bias.x; awy += bias.y; awz += bias.z; aww += bias.w;
  abx += bias.x; aby += bias.y; abz += bias.z; abw += bias.w;

  const float p = pov[n];
  const float q = 1.0f - p;

  float4 o1, o2;  // o1 = cols [0,768): p*w + q*b ; o2 = cols [768,1536): p*b + q*w
  o1.x = fmaxf(p * awx + q * abx, 0.f);
  o1.y = fmaxf(p * awy + q * aby, 0.f);
  o1.z = fmaxf(p * awz + q * abz, 0.f);
  o1.w = fmaxf(p * aww + q * abw, 0.f);
  o2.x = fmaxf(p * abx + q * awx, 0.f);
  o2.y = fmaxf(p * aby + q * awy, 0.f);
  o2.z = fmaxf(p * abz + q * awz, 0.f);
  o2.w = fmaxf(p * abw + q * aww, 0.f);

  *(float4*)&base[(size_t)n * TWOB + c]        = o1;
  *(float4*)&base[(size_t)n * TWOB + BASE + c] = o2;
}

// ------------------------------ Kernel 2 -----------------------------------
// grid = N/16 blocks, one wave32 per block; each wave owns a 16-row tile.
// fc0 GEMM via V_WMMA_F32_16X16X4_F32, K=1536 consumed in 12 chunks of 128.
__global__ __launch_bounds__(32) void nnue_head(
    const float* __restrict__ base,
    const float* __restrict__ fc0_w, const float* __restrict__ fc0_b,
    const float* __restrict__ fc1_w, const float* __restrict__ fc1_b,
    const float* __restrict__ fc2_w, const float* __restrict__ fc2_b,
    const float* __restrict__ fc3_w, const float* __restrict__ fc3_b,
    float*       __restrict__ out)
{
  __shared__ float As[16][132];   // 16x128 A chunk, padded (stride 132 -> no 16-way bank conflict)
  __shared__ float Bs[128][16];   // 128x16 B chunk; cols 8..15 zero-padded
  __shared__ float Xs[16][24];    // per-row activations [x0 | x1 | x2]

  const int    lane = threadIdx.x;                 // 0..31, full wave active
  const size_t row0 = (size_t)blockIdx.x * 16;

  v8f acc = {0.f, 0.f, 0.f, 0.f, 0.f, 0.f, 0.f, 0.f};

  // WMMA operand addressing (ISA 7.12.2, f32 16x16x4):
  //  A: lanes 0-15 -> M=lane, K=0(.x)/1(.y); lanes 16-31 -> M=lane-16, K=2/3
  //  B: lane provides column N=lane&15 at the same K pair
  const int m   = lane & 15;
  const int kof = (lane >> 4) * 2;
  const int nb  = lane & 15;

  for (int ch = 0; ch < 12; ++ch) {
    const int k0 = ch * 128;
    // stage A chunk: 2048 elems, coalesced 128-col rows
    for (int i = lane; i < 16 * 128; i += 32) {
      const int r  = i >> 7;
      const int cc = i & 127;
      As[r][cc] = base[(row0 + r) * TWOB + k0 + cc];
    }
    // stage B chunk: fc0_w is [8,1536]; zero-pad cols 8..15
    for (int i = lane; i < 128 * 16; i += 32) {
      const int kk = i >> 4;
      const int nn = i & 15;
      Bs[kk][nn] = (nn < 8) ? fc0_w[nn * TWOB + k0 + kk] : 0.0f;
    }
    __syncthreads();

    #pragma unroll 4
    for (int ks = 0; ks < 128; ks += 4) {
      v2f a, b;
      a.x = As[m][ks + kof];
      a.y = As[m][ks + kof + 1];
      b.x = Bs[ks + kof][nb];
      b.y = Bs[ks + kof + 1][nb];
      // D = A(16x4) x B(4x16) + C ; full f32 matrix pipe
      acc = __builtin_amdgcn_wmma_f32_16x16x4_f32(
          false, a, false, b, (short)0, acc, false, false);
    }
    __syncthreads();
  }

  // Epilogue: D layout -> VGPR g holds (M = g + 8*(lane>=16), N = lane&15)
  {
    const int ncol  = lane & 15;
    const int rhalf = (lane >> 4) * 8;
    if (ncol < 8) {
      const float b0 = fc0_b[ncol];
      #pragma unroll
      for (int g = 0; g < 8; ++g) {
        Xs[rhalf + g][ncol] = fmaxf(acc[g] + b0, 0.0f);
      }
    }
  }
  __syncthreads();

  // Tiny head: fc1 (8->8), fc2 (16->8), fc3 (24->1); one lane per row.
  if (lane < 16) {
    const int r = lane;
    #pragma unroll
    for (int o = 0; o < 8; ++o) {
      float s = fc1_b[o];
      #pragma unroll
      for (int i = 0; i < 8; ++i) s += fc1_w[o * 8 + i] * Xs[r][i];
      Xs[r][8 + o] = fmaxf(s, 0.0f);
    }
    #pragma unroll
    for (int o = 0; o < 8; ++o) {
      float s = fc2_b[o];
      #pragma unroll
      for (int i = 0; i < 16; ++i) s += fc2_w[o * 16 + i] * Xs[r][i];
      Xs[r][16 + o] = fmaxf(s, 0.0f);
    }
    float s = fc3_b[0];
    #pragma unroll
    for (int i = 0; i < 24; ++i) s += fc3_w[i] * Xs[r][i];
    out[row0 + r] = s;
  }
}

// ------------------------------ Launcher -----------------------------------
extern "C" void kernel_launch(void* const* d_in, const int* in_sizes, int n_in,
                              void* d_out, int out_size, void* d_ws, size_t ws_size,
                              hipStream_t stream) {
  const float* pov        = (const float*)d_in[0];
  const int*   widx       = (const int*)  d_in[1];
  const int*   bidx       = (const int*)  d_in[2];
  const float* affine_w   = (const float*)d_in[3];
  const float* affine_b   = (const float*)d_in[4];
  const float* factored_w = (const float*)d_in[5];
  const float* fc0_w      = (const float*)d_in[6];
  const float* fc0_b      = (const float*)d_in[7];
  const float* fc1_w      = (const float*)d_in[8];
  const float* fc1_b      = (const float*)d_in[9];
  const float* fc2_w      = (const float*)d_in[10];
  const float* fc2_b      = (const float*)d_in[11];
  const float* fc3_w      = (const float*)d_in[12];
  const float* fc3_b      = (const float*)d_in[13];
  float*       out        = (float*)d_out;
  float*       basews     = (float*)d_ws;   // N*1536 floats = 25.2 MB scratch

  nnue_ft_blend<<<NN, 192, 0, stream>>>(pov, widx, bidx,
                                        affine_w, affine_b, factored_w, basews);
  nnue_head<<<NN / 16, 32, 0, stream>>>(basews, fc0_w, fc0_b, fc1_w, fc1_b,
                                        fc2_w, fc2_b, fc3_w, fc3_b, out);
}